// ResGATGIN_Encoder_34926674051128
// MI455X (gfx1250) — compile-verified
//
#include <hip/hip_runtime.h>
#include <math.h>

// Problem constants (match reference)
#define NN 50000
#define EE 800000
#define HH 4
// C_IN=128, H*C_HID=256, C_HID=C_OUT=64
#define WPAD 72   // padded LDS row stride (floats): 2*72 % 64 == 16 -> halves hit disjoint banks

typedef __attribute__((ext_vector_type(2))) float v2f;
typedef __attribute__((ext_vector_type(8))) float v8f;

__device__ __forceinline__ v8f wmma4(v2f a, v2f b, v8f c) {
  // V_WMMA_F32_16X16X4_F32 : D = A(16x4) * B(4x16) + C(16x16)
  return __builtin_amdgcn_wmma_f32_16x16x4_f32(false, a, false, b, (short)0, c,
                                               false, false);
}

__device__ __forceinline__ void atomicMaxF(float* addr, float value) {
  if (value >= 0.0f)
    atomicMax((int*)addr, __float_as_int(value));
  else
    atomicMin((unsigned int*)addr, __float_as_uint(value));
}

// Stage a 64x64 f32 weight matrix into padded LDS with CDNA5 async LDS loads.
// 256 threads; each issues 4 x b128 async copies, then waits its ASYNCcnt.
__device__ __forceinline__ void stage_w64(const float* __restrict__ W,
                                          float* __restrict__ wlds) {
  const int tflat = threadIdx.x;  // 0..255
#pragma unroll
  for (int i = 0; i < 4; ++i) {
    int ci = tflat + i * 256;      // 1024 float4 chunks
    int row = ci >> 4;
    int col = (ci & 15) * 4;
    unsigned ldsoff = (unsigned)(uintptr_t)(wlds + WPAD * row + col);
    const float* gp = W + row * 64 + col;
    asm volatile("global_load_async_to_lds_b128 %0, %1, off"
                 :: "v"(ldsoff), "v"(gp) : "memory");
  }
  asm volatile("s_wait_asynccnt 0x0" ::: "memory");
  __syncthreads();
}

// ---------------------------------------------------------------- init
__global__ void k_init(float4* __restrict__ agg, float4* __restrict__ den,
                       float4* __restrict__ gin, float4* __restrict__ emax) {
  int t = blockIdx.x * blockDim.x + threadIdx.x;
  const float4 z = make_float4(0.f, 0.f, 0.f, 0.f);
  const float ninf = -3.402823466e38f;
  if (t < NN * 64) agg[t] = z;                    // N*256 floats
  if (t < NN * 16) gin[t] = z;                    // N*64 floats
  if (t < NN) { den[t] = z; emax[t] = make_float4(ninf, ninf, ninf, ninf); }
}

// ------------------------------------------- GEMM1: xh = x @ Wg  (WMMA f32)
// x: [N,128]; Wg: [128,256]; xh: [N,256]
// block = 128 (4 waves); grid = N/16. Each wave: one 16-row strip x 64 cols
// (4 independent accumulators -> 4 parallel WMMA chains, A reused 4x).
__global__ void k_gemm_xh(const float* __restrict__ x,
                          const float* __restrict__ Wg,
                          float* __restrict__ xh) {
  const int lane = threadIdx.x & 31;
  const int wave = threadIdx.x >> 5;       // 0..3 -> col base wave*64
  const int row0 = blockIdx.x * 16;
  const int cbase = wave * 64;
  const int lm = lane & 15;
  const int lh = lane >> 4;

  v8f acc0 = {}, acc1 = {}, acc2 = {}, acc3 = {};
  const float* xr = x + (size_t)(row0 + lm) * 128;
#pragma unroll
  for (int k = 0; k < 128; k += 4) {
    v2f a;
    a.x = xr[k + 2 * lh + 0];
    a.y = xr[k + 2 * lh + 1];
    const float* w0 = Wg + (size_t)(k + 2 * lh) * 256 + cbase + lm;
    const float* w1 = w0 + 256;  // next K row
    v2f b0 = {w0[0],  w1[0]};
    v2f b1 = {w0[16], w1[16]};
    v2f b2 = {w0[32], w1[32]};
    v2f b3 = {w0[48], w1[48]};
    acc0 = wmma4(a, b0, acc0);
    acc1 = wmma4(a, b1, acc1);
    acc2 = wmma4(a, b2, acc2);
    acc3 = wmma4(a, b3, acc3);
  }
  float* o = xh + (size_t)row0 * 256 + cbase + lm;
#pragma unroll
  for (int j = 0; j < 8; ++j) {
    size_t r = (size_t)(j + 8 * lh) * 256;
    o[r +  0] = acc0[j];
    o[r + 16] = acc1[j];
    o[r + 32] = acc2[j];
    o[r + 48] = acc3[j];
  }
}

// --------------------------------- attention scores a_src/a_dst : [N,H]
__global__ void k_att_scores(const float* __restrict__ xh,
                             const float* __restrict__ att_src,
                             const float* __restrict__ att_dst,
                             float* __restrict__ a_s, float* __restrict__ a_d) {
  int t = blockIdx.x * blockDim.x + threadIdx.x;  // t = n*H + h
  if (t >= NN * HH) return;
  int h = t & 3;
  int n = t >> 2;
  const float4* v4 = (const float4*)(xh + (size_t)n * 256 + h * 64);
  const float4* s4 = (const float4*)(att_src + h * 64);
  const float4* d4 = (const float4*)(att_dst + h * 64);
  float ss = 0.0f, sd = 0.0f;
#pragma unroll
  for (int j = 0; j < 16; ++j) {
    float4 xv = v4[j], as = s4[j], ad = d4[j];
    ss += xv.x * as.x + xv.y * as.y + xv.z * as.z + xv.w * as.w;
    sd += xv.x * ad.x + xv.y * ad.y + xv.z * ad.z + xv.w * ad.w;
  }
  a_s[t] = ss;
  a_d[t] = sd;
}

// ------------------------------------- edge pass 1: segment max (leaky relu)
__global__ void k_edge_max(const int* __restrict__ src,
                           const int* __restrict__ dst,
                           const float* __restrict__ a_s,
                           const float* __restrict__ a_d,
                           float* __restrict__ emax) {
  int e = blockIdx.x * blockDim.x + threadIdx.x;
  if (e >= EE + NN) return;
  int s = (e < EE) ? src[e] : (e - EE);
  int d = (e < EE) ? dst[e] : (e - EE);
  float4 vs = *(const float4*)(a_s + (size_t)s * 4);
  float4 vd = *(const float4*)(a_d + (size_t)d * 4);
  float v[4] = {vs.x + vd.x, vs.y + vd.y, vs.z + vd.z, vs.w + vd.w};
#pragma unroll
  for (int h = 0; h < HH; ++h) {
    float w = (v[h] > 0.0f) ? v[h] : 0.2f * v[h];  // leaky_relu
    atomicMaxF(&emax[d * 4 + h], w);
  }
}

// ---------------------- edge pass 2: p = exp(e - emax[d]); denom += p
__global__ void k_edge_exp(const int* __restrict__ src,
                           const int* __restrict__ dst,
                           const float* __restrict__ a_s,
                           const float* __restrict__ a_d,
                           const float* __restrict__ emax,
                           float* __restrict__ p, float* __restrict__ den) {
  int e = blockIdx.x * blockDim.x + threadIdx.x;
  if (e >= EE + NN) return;
  int s = (e < EE) ? src[e] : (e - EE);
  int d = (e < EE) ? dst[e] : (e - EE);
  float4 vs = *(const float4*)(a_s + (size_t)s * 4);
  float4 vd = *(const float4*)(a_d + (size_t)d * 4);
  float4 mx = *(const float4*)(emax + (size_t)d * 4);
  float v[4] = {vs.x + vd.x, vs.y + vd.y, vs.z + vd.z, vs.w + vd.w};
  float m[4] = {mx.x, mx.y, mx.z, mx.w};
  float4 pe;
  float* pp = (float*)&pe;
#pragma unroll
  for (int h = 0; h < HH; ++h) {
    float w = (v[h] > 0.0f) ? v[h] : 0.2f * v[h];
    pp[h] = __expf(w - m[h]);
    atomicAdd(&den[d * 4 + h], pp[h]);
  }
  *(float4*)(p + (size_t)e * 4) = pe;
}

// ------------------------------- normalize: p <- p / (denom[d] + 1e-16)
__global__ void k_alpha_norm(const int* __restrict__ dst,
                             const float* __restrict__ den,
                             float* __restrict__ p) {
  int t = blockIdx.x * blockDim.x + threadIdx.x;  // t = e*H + h
  if (t >= (EE + NN) * HH) return;
  int e = t >> 2;
  int h = t & 3;
  int d = (e < EE) ? dst[e] : (e - EE);
  p[t] = p[t] / (den[d * 4 + h] + 1e-16f);
}

// ------------- edge pass 3: agg[d,h,c] += alpha[e,h] * xh[s,h,c]  (heavy)
// thread = (edge, float4-chunk); float4 B128 reads, coalesced f32 atomics.
__global__ void k_edge_agg(const int* __restrict__ src,
                           const int* __restrict__ dst,
                           const float* __restrict__ p,
                           const float* __restrict__ xh,
                           float* __restrict__ agg) {
  int t = blockIdx.x * blockDim.x + threadIdx.x;  // t = e*16 + c4
  if (t >= (EE + NN) * 16) return;
  int e = t >> 4;
  int c = (t & 15) * 4;
  int s = (e < EE) ? src[e] : (e - EE);
  int d = (e < EE) ? dst[e] : (e - EE);
  float4 al = *(const float4*)(p + (size_t)e * 4);
  float a[4] = {al.x, al.y, al.z, al.w};
  const float* xs = xh + (size_t)s * 256 + c;
  float* ad = agg + (size_t)d * 256 + c;
#pragma unroll
  for (int h = 0; h < HH; ++h) {
    float4 v = *(const float4*)(xs + h * 64);
    atomicAdd(ad + h * 64 + 0, a[h] * v.x);
    atomicAdd(ad + h * 64 + 1, a[h] * v.y);
    atomicAdd(ad + h * 64 + 2, a[h] * v.z);
    atomicAdd(ad + h * 64 + 3, a[h] * v.w);
  }
}

// -------------------- GAT epilogue: head mean + bias + BN + ReLU -> x_gat
__global__ void k_gat_post(const float* __restrict__ agg,
                           const float* __restrict__ gat_bias,
                           const float* __restrict__ g, const float* __restrict__ b,
                           const float* __restrict__ m, const float* __restrict__ v,
                           float* __restrict__ xg) {
  int t = blockIdx.x * blockDim.x + threadIdx.x;  // t = n*16 + c4
  if (t >= NN * 16) return;
  int n = t >> 4;
  int c = (t & 15) * 4;
  const float* an = agg + (size_t)n * 256 + c;
  float4 s0 = *(const float4*)(an + 0);
  float4 s1 = *(const float4*)(an + 64);
  float4 s2 = *(const float4*)(an + 128);
  float4 s3 = *(const float4*)(an + 192);
  float4 gb = *(const float4*)(gat_bias + c);
  float4 gg = *(const float4*)(g + c);
  float4 bb = *(const float4*)(b + c);
  float4 mm = *(const float4*)(m + c);
  float4 vv = *(const float4*)(v + c);
  float4 r;
  r.x = fmaxf(gg.x * (0.25f * (s0.x + s1.x + s2.x + s3.x) + gb.x - mm.x) * rsqrtf(vv.x + 1e-5f) + bb.x, 0.f);
  r.y = fmaxf(gg.y * (0.25f * (s0.y + s1.y + s2.y + s3.y) + gb.y - mm.y) * rsqrtf(vv.y + 1e-5f) + bb.y, 0.f);
  r.z = fmaxf(gg.z * (0.25f * (s0.z + s1.z + s2.z + s3.z) + gb.z - mm.z) * rsqrtf(vv.z + 1e-5f) + bb.z, 0.f);
  r.w = fmaxf(gg.w * (0.25f * (s0.w + s1.w + s2.w + s3.w) + gb.w - mm.w) * rsqrtf(vv.w + 1e-5f) + bb.w, 0.f);
  *(float4*)(xg + (size_t)n * 64 + c) = r;
}

// ------------------------------- GIN neighbor scatter (original edges only)
__global__ void k_gin_scatter(const int* __restrict__ src,
                              const int* __restrict__ dst,
                              const float* __restrict__ xg,
                              float* __restrict__ gin) {
  int t = blockIdx.x * blockDim.x + threadIdx.x;  // t = e*16 + c4
  if (t >= EE * 16) return;
  int e = t >> 4;
  int c = (t & 15) * 4;
  float4 v = *(const float4*)(xg + (size_t)src[e] * 64 + c);
  float* gd = gin + (size_t)dst[e] * 64 + c;
  atomicAdd(gd + 0, v.x);
  atomicAdd(gd + 1, v.y);
  atomicAdd(gd + 2, v.z);
  atomicAdd(gd + 3, v.w);
}

// ------------------------------- gin <- (1+eps)*xg + gin
__global__ void k_gin_combine(const float* __restrict__ eps,
                              const float4* __restrict__ xg,
                              float4* __restrict__ gin) {
  int t = blockIdx.x * blockDim.x + threadIdx.x;
  if (t >= NN * 16) return;
  float e1 = 1.0f + eps[0];
  float4 a = xg[t], b = gin[t];
  gin[t] = make_float4(e1 * a.x + b.x, e1 * a.y + b.y, e1 * a.z + b.z, e1 * a.w + b.w);
}

// --------------- MLP1: h1 = relu(bn1(gin @ W1 + b1))  (WMMA f32, fused BN)
// block = 256 (8 waves); W1 staged in LDS via async-LDS copies.
// Each wave: its own 16-row strip x all 64 cols (4 accumulators).
__global__ void k_mlp1(const float* __restrict__ gin, const float* __restrict__ W1,
                       const float* __restrict__ b1,
                       const float* __restrict__ g, const float* __restrict__ b,
                       const float* __restrict__ m, const float* __restrict__ v,
                       float* __restrict__ h1) {
  __shared__ float wlds[64 * WPAD];
  stage_w64(W1, wlds);

  const int lane = threadIdx.x & 31;
  const int wave = threadIdx.x >> 5;
  const int row0 = (blockIdx.x * 8 + wave) * 16;
  const int lm = lane & 15;
  const int lh = lane >> 4;
  if (row0 < NN) {  // wave-uniform: EXEC stays all-ones for WMMA
    v8f acc0 = {}, acc1 = {}, acc2 = {}, acc3 = {};
    const float* ar = gin + (size_t)(row0 + lm) * 64;
#pragma unroll
    for (int k = 0; k < 64; k += 4) {
      v2f a;
      a.x = ar[k + 2 * lh + 0];
      a.y = ar[k + 2 * lh + 1];
      const float* w0 = wlds + (k + 2 * lh) * WPAD + lm;  // ds_load
      const float* w1 = w0 + WPAD;
      v2f b0 = {w0[0],  w1[0]};
      v2f b1_ = {w0[16], w1[16]};
      v2f b2 = {w0[32], w1[32]};
      v2f b3 = {w0[48], w1[48]};
      acc0 = wmma4(a, b0, acc0);
      acc1 = wmma4(a, b1_, acc1);
      acc2 = wmma4(a, b2, acc2);
      acc3 = wmma4(a, b3, acc3);
    }
#pragma unroll
    for (int tile = 0; tile < 4; ++tile) {
      v8f acc = (tile == 0) ? acc0 : (tile == 1) ? acc1 : (tile == 2) ? acc2 : acc3;
      int col = tile * 16 + lm;
      float scale = g[col] * rsqrtf(v[col] + 1e-5f);
      float off = b[col] - m[col] * scale;
      float bias = b1[col];
#pragma unroll
      for (int j = 0; j < 8; ++j) {
        float val = (acc[j] + bias) * scale + off;
        h1[(size_t)(row0 + j + 8 * lh) * 64 + col] = fmaxf(val, 0.0f);
      }
    }
  }
}

// --------- MLP2: out = relu(h1 @ W2 + b2 + x_gat)  (WMMA f32, fused residual)
__global__ void k_mlp2(const float* __restrict__ h1, const float* __restrict__ W2,
                       const float* __restrict__ b2, const float* __restrict__ xg,
                       float* __restrict__ out) {
  __shared__ float wlds[64 * WPAD];
  stage_w64(W2, wlds);

  const int lane = threadIdx.x & 31;
  const int wave = threadIdx.x >> 5;
  const int row0 = (blockIdx.x * 8 + wave) * 16;
  const int lm = lane & 15;
  const int lh = lane >> 4;
  if (row0 < NN) {
    v8f acc0 = {}, acc1 = {}, acc2 = {}, acc3 = {};
    const float* ar = h1 + (size_t)(row0 + lm) * 64;
#pragma unroll
    for (int k = 0; k < 64; k += 4) {
      v2f a;
      a.x = ar[k + 2 * lh + 0];
      a.y = ar[k + 2 * lh + 1];
      const float* w0 = wlds + (k + 2 * lh) * WPAD + lm;
      const float* w1 = w0 + WPAD;
      v2f b0 = {w0[0],  w1[0]};
      v2f b1_ = {w0[16], w1[16]};
      v2f b2_ = {w0[32], w1[32]};
      v2f b3 = {w0[48], w1[48]};
      acc0 = wmma4(a, b0, acc0);
      acc1 = wmma4(a, b1_, acc1);
      acc2 = wmma4(a, b2_, acc2);
      acc3 = wmma4(a, b3, acc3);
    }
#pragma unroll
    for (int tile = 0; tile < 4; ++tile) {
      v8f acc = (tile == 0) ? acc0 : (tile == 1) ? acc1 : (tile == 2) ? acc2 : acc3;
      int col = tile * 16 + lm;
      float bias = b2[col];
#pragma unroll
      for (int j = 0; j < 8; ++j) {
        int row = row0 + j + 8 * lh;
        float val = acc[j] + bias + xg[(size_t)row * 64 + col];
        out[(size_t)row * 64 + col] = fmaxf(val, 0.0f);
      }
    }
  }
}

extern "C" void kernel_launch(void* const* d_in, const int* in_sizes, int n_in,
                              void* d_out, int out_size, void* d_ws, size_t ws_size,
                              hipStream_t stream) {
  (void)in_sizes; (void)n_in; (void)out_size; (void)ws_size;

  const float* x        = (const float*)d_in[0];
  const int*   eidx     = (const int*)  d_in[1];
  const float* Wg       = (const float*)d_in[2];
  const float* att_src  = (const float*)d_in[3];
  const float* att_dst  = (const float*)d_in[4];
  const float* gat_bias = (const float*)d_in[5];
  const float* bn_g     = (const float*)d_in[6];
  const float* bn_b     = (const float*)d_in[7];
  const float* bn_m     = (const float*)d_in[8];
  const float* bn_v     = (const float*)d_in[9];
  const float* eps_gin  = (const float*)d_in[10];
  const float* W1       = (const float*)d_in[11];
  const float* b1       = (const float*)d_in[12];
  const float* bn1_g    = (const float*)d_in[13];
  const float* bn1_b    = (const float*)d_in[14];
  const float* bn1_m    = (const float*)d_in[15];
  const float* bn1_v    = (const float*)d_in[16];
  const float* W2       = (const float*)d_in[17];
  const float* b2       = (const float*)d_in[18];

  const int* src = eidx;       // edge_index[0]
  const int* dst = eidx + EE;  // edge_index[1]

  // workspace layout (floats)
  float* ws   = (float*)d_ws;
  float* xh   = ws;                               // N*256
  float* a_s  = xh   + (size_t)NN * 256;          // N*4
  float* a_d  = a_s  + (size_t)NN * 4;            // N*4
  float* emax = a_d  + (size_t)NN * 4;            // N*4
  float* p    = emax + (size_t)NN * 4;            // (E+N)*4
  float* den  = p    + (size_t)(EE + NN) * 4;     // N*4
  float* agg  = den  + (size_t)NN * 4;            // N*256
  float* xg   = agg  + (size_t)NN * 256;          // N*64
  float* gin  = xg   + (size_t)NN * 64;           // N*64
  float* h1   = gin  + (size_t)NN * 64;           // N*64

  float* out = (float*)d_out;

  const int B = 256;
  // 0) re-init accumulators every call (atomic kernels accumulate)
  k_init<<<(NN * 64 + B - 1) / B, B, 0, stream>>>((float4*)agg, (float4*)den,
                                                  (float4*)gin, (float4*)emax);
  // 1) xh = x @ Wg   (50000x128x256, fp32 WMMA, 4 tiles/wave)
  k_gemm_xh<<<NN / 16, 128, 0, stream>>>(x, Wg, xh);
  // 2) attention scores
  k_att_scores<<<(NN * HH + B - 1) / B, B, 0, stream>>>(xh, att_src, att_dst, a_s, a_d);
  // 3) segment max over incoming edges
  k_edge_max<<<(EE + NN + B - 1) / B, B, 0, stream>>>(src, dst, a_s, a_d, emax);
  // 4) exp + denom
  k_edge_exp<<<(EE + NN + B - 1) / B, B, 0, stream>>>(src, dst, a_s, a_d, emax, p, den);
  // 5) alpha = p / denom
  k_alpha_norm<<<((EE + NN) * HH + B - 1) / B, B, 0, stream>>>(dst, den, p);
  // 6) weighted scatter aggregation (dominant cost; L2-resident atomics)
  k_edge_agg<<<((EE + NN) * 16 + B - 1) / B, B, 0, stream>>>(src, dst, p, xh, agg);
  // 7) head mean + bias + BN + ReLU
  k_gat_post<<<(NN * 16 + B - 1) / B, B, 0, stream>>>(agg, gat_bias, bn_g, bn_b, bn_m, bn_v, xg);
  // 8) GIN neighbor sum
  k_gin_scatter<<<(EE * 16 + B - 1) / B, B, 0, stream>>>(src, dst, xg, gin);
  // 9) (1+eps)*x + agg
  k_gin_combine<<<(NN * 16 + B - 1) / B, B, 0, stream>>>(eps_gin, (const float4*)xg,
                                                         (float4*)gin);
  // 10) MLP layer 1 (WMMA + async-LDS weight staging + fused BN/ReLU)
  k_mlp1<<<(NN / 16 + 7) / 8, 256, 0, stream>>>(gin, W1, b1, bn1_g, bn1_b, bn1_m, bn1_v, h1);
  // 11) MLP layer 2 (WMMA + async-LDS weight staging + fused bias/residual/ReLU)
  k_mlp2<<<(NN / 16 + 7) / 8, 256, 0, stream>>>(h1, W2, b2, xg, out);
}